// GLGCM_Loss_71159018160693
// MI455X (gfx1250) — compile-verified
//
#include <hip/hip_runtime.h>
#include <hip/hip_bf16.h>

typedef __attribute__((ext_vector_type(2))) float v2f;
typedef __attribute__((ext_vector_type(8))) float v8f;

#define NB   256
#define IMW  1024
#define IMH  1024
#define BATCH 8
#define NIN   4
#define WPI   16                    // workgroups per image in histogram phase
#define ROWS_PER_WG (IMH / WPI)     // 64 rows per workgroup
// per-image histogram sum: every one of H*(W-1) pairs contributes twice
#define PAIR_SUM 2095104.0          // 2 * 1024 * 1023

// ---------------------------------------------------------------------------
// zero the histogram workspace (graph-replay safe: re-zeroed every launch)
// ---------------------------------------------------------------------------
__global__ void zero_hist_kernel(unsigned int* __restrict__ hist, int n) {
    int i = blockIdx.x * blockDim.x + threadIdx.x;
    int stride = gridDim.x * blockDim.x;
    for (; i < n; i += stride) hist[i] = 0u;
}

// ---------------------------------------------------------------------------
// Sobel -> bin -> per-image GLCM histogram. One 256KB LDS histogram per
// workgroup (CDNA5: 320KB LDS per WGP), flushed once via global atomics.
// ---------------------------------------------------------------------------
__device__ __forceinline__ void loadcol(const float* __restrict__ im, int y, int x,
                                        float& r0, float& r1, float& r2) {
    if ((unsigned)x < (unsigned)IMW) {
        r0 = (y > 0)       ? im[(y - 1) * IMW + x] : 0.f;
        r1 =                 im[y * IMW + x];
        r2 = (y < IMH - 1) ? im[(y + 1) * IMW + x] : 0.f;
    } else {
        r0 = r1 = r2 = 0.f;
    }
}

__device__ __forceinline__ int bin_of(float a0, float a1, float a2,
                                      float b0, float b1, float b2,
                                      float c0, float c1, float c2) {
    // cross-correlation with KX/KY, zero padding (cols a=x-1, b=x, c=x+1; rows 0=y-1,2=y+1)
    float gx = (c0 + 2.f * c1 + c2) - (a0 + 2.f * a1 + a2);
    float gy = (a2 + 2.f * b2 + c2) - (a0 + 2.f * b0 + c0);
    float g  = sqrtf(gx * gx + gy * gy);
    return ((int)(g * 255.0f)) & (NB - 1);   // trunc (g>=0) then mod 256 == byte cast
}

__global__ __launch_bounds__(256) void hist_kernel(
        const float* __restrict__ in0, const float* __restrict__ in1,
        const float* __restrict__ in2, const float* __restrict__ in3,
        unsigned int* __restrict__ ghist) {
    __shared__ unsigned int lh[NB * NB];     // 256 KB private GLCM (fits CDNA5 LDS)

    const int t  = blockIdx.y;               // which input tensor
    const int b  = blockIdx.x / WPI;         // image in batch
    const int wg = blockIdx.x % WPI;         // row-block within image
    const float* base = (t == 0) ? in0 : (t == 1) ? in1 : (t == 2) ? in2 : in3;
    const float* im = base + (size_t)b * (IMW * IMH);

    for (int i = threadIdx.x; i < NB * NB; i += blockDim.x) lh[i] = 0u;
    __syncthreads();

    const int tid = threadIdx.x;
    const int y   = wg * ROWS_PER_WG + (tid >> 2);   // 64 rows, 4 segments/row
    const int seg = tid & 3;
    const int xs  = seg * 256;                        // first pair index
    const int xe  = min(xs + 255, IMW - 2);           // last pair index (x, x+1)

    // sliding 3-column window
    float a0, a1, a2, b0, b1, b2, c0, c1, c2;
    loadcol(im, y, xs - 1, a0, a1, a2);
    loadcol(im, y, xs,     b0, b1, b2);
    loadcol(im, y, xs + 1, c0, c1, c2);
    int prev = bin_of(a0, a1, a2, b0, b1, b2, c0, c1, c2);

    for (int x = xs; x <= xe; ++x) {
        a0 = b0; a1 = b1; a2 = b2;
        b0 = c0; b1 = c1; b2 = c2;
        loadcol(im, y, x + 2, c0, c1, c2);
        int cur = bin_of(a0, a1, a2, b0, b1, b2, c0, c1, c2);
        atomicAdd(&lh[prev * NB + cur], 1u);   // one-sided; symmetrized in reduction
        prev = cur;
    }
    __syncthreads();

    unsigned int* gh = ghist + (((size_t)(t * BATCH + b)) << 16);
    for (int i = threadIdx.x; i < NB * NB; i += blockDim.x) {
        unsigned int v = lh[i];
        if (v) atomicAdd(&gh[i], v);
    }
}

// ---------------------------------------------------------------------------
// Per (tensor, image, 16-row strip): WMMA f32 16x16x4 reduction of the
// symmetrized GLCM.  A = Hs tile (documented layout), B = all-ones (layout
// agnostic), giving row sums; second chain pre-scales A by global k to get
// j-weighted row sums.  Entropy/energy accumulated elementwise in the loop.
// ---------------------------------------------------------------------------
__global__ __launch_bounds__(32) void moment_kernel(
        const unsigned int* __restrict__ ghist, float* __restrict__ partials) {
    const int id    = blockIdx.x;            // 0..511 = (t*8 + b)*16 + strip
    const int t     = id >> 7;
    const int b     = (id >> 4) & 7;
    const int strip = id & 15;
    const int R0    = strip * 16;
    const unsigned int* Hc = ghist + (((size_t)(t * BATCH + b)) << 16);

    const int lane = threadIdx.x;
    const int half = lane >> 4;              // A layout: lanes 16..31 hold K+2,K+3
    const int mr   = lane & 15;
    const int row  = R0 + mr;
    const float Sinv = (float)(1.0 / PAIR_SUM);

    v8f acc1 = {0.f, 0.f, 0.f, 0.f, 0.f, 0.f, 0.f, 0.f};
    v8f acc2 = {0.f, 0.f, 0.f, 0.f, 0.f, 0.f, 0.f, 0.f};
    v2f ones = {1.f, 1.f};
    float e2 = 0.f, ent = 0.f;

    for (int K0 = 0; K0 < NB; K0 += 4) {
        const int k0 = K0 + half * 2;
        const int k1 = k0 + 1;
        // symmetrized count Hs[row][k] = h[row][k] + h[k][row]
        float h0 = (float)(Hc[row * NB + k0] + Hc[k0 * NB + row]);
        float h1 = (float)(Hc[row * NB + k1] + Hc[k1 * NB + row]);
        v2f a1 = {h0, h1};
        v2f a2 = {h0 * (float)k0, h1 * (float)k1};   // j-weighted A chunk
        acc1 = __builtin_amdgcn_wmma_f32_16x16x4_f32(false, a1, false, ones,
                                                     (short)0, acc1, false, false);
        acc2 = __builtin_amdgcn_wmma_f32_16x16x4_f32(false, a2, false, ones,
                                                     (short)0, acc2, false, false);
        float p0 = h0 * Sinv, p1 = h1 * Sinv;
        e2  += p0 * p0 + p1 * p1;
        ent += p0 * logf(p0 + 1e-8f) + p1 * logf(p1 + 1e-8f);
    }

    // C/D layout: VGPR r holds row M=r (lanes 0-15) / M=r+8 (lanes 16-31);
    // columns identical (B==ones) -> 16-fold replication divided out below.
    float mi = 0.f, m2 = 0.f, mij = 0.f;
#pragma unroll
    for (int r = 0; r < 8; ++r) {
        float gm = (float)(R0 + r + half * 8);
        float rs = acc1[r];
        mi  += gm * rs;
        m2  += gm * gm * rs;
        mij += gm * acc2[r];
    }
    for (int off = 16; off; off >>= 1) {
        mi  += __shfl_xor(mi,  off);
        m2  += __shfl_xor(m2,  off);
        mij += __shfl_xor(mij, off);
        e2  += __shfl_xor(e2,  off);
        ent += __shfl_xor(ent, off);
    }
    if (lane == 0) {
        float* p = partials + id * 8;
        p[0] = e2;                   // sum P^2                (already normalized)
        p[1] = ent;                  // sum P*log(P+eps)
        p[2] = mi  * (1.f / 16.f);   // raw  sum i  * Hs
        p[3] = m2  * (1.f / 16.f);   // raw  sum i^2* Hs
        p[4] = mij * (1.f / 16.f);   // raw  sum i*j* Hs
    }
}

// ---------------------------------------------------------------------------
// Deterministic final combine in double precision, single thread.
// ---------------------------------------------------------------------------
__global__ void final_kernel(const float* __restrict__ partials,
                             float* __restrict__ out) {
    if (threadIdx.x != 0 || blockIdx.x != 0) return;
    const double S = PAIR_SUM;
    double feat[NIN][3];
    for (int t = 0; t < NIN; ++t) {
        double E2 = 0, ENT = 0, MI = 0, M2 = 0, MIJ = 0;
        for (int s = 0; s < BATCH * 16; ++s) {
            const float* p = partials + (t * BATCH * 16 + s) * 8;
            E2 += p[0]; ENT += p[1]; MI += p[2]; M2 += p[3]; MIJ += p[4];
        }
        double m   = MI  / S;        // mean_i == mean_j (Hs symmetric)
        double m2j = M2  / S;        // == sum j^2 * P via symmetry
        double mij = MIJ / S;
        // num   = sum P (mi - i)(mj - j) = Mij + 6 m^2      (sum P = 8)
        // std_i = std_j = sqrt(8m^2 - 2m^2 + M2j) -> denom = 6 m^2 + M2j
        double num   = mij + 6.0 * m * m;
        double denom = 6.0 * m * m + m2j;
        feat[t][0] = E2;
        feat[t][1] = (denom == 0.0) ? 0.0 : num / denom;
        feat[t][2] = -ENT;
    }
    double loss = fabs(feat[0][0] - feat[1][0]) + fabs(feat[0][1] - feat[1][1])
                + fabs(feat[0][2] - feat[1][2]) + fabs(feat[2][0] - feat[3][0])
                + fabs(feat[2][1] - feat[3][1]) + fabs(feat[2][2] - feat[3][2]);
    out[0] = (float)loss;
}

// ---------------------------------------------------------------------------
extern "C" void kernel_launch(void* const* d_in, const int* in_sizes, int n_in,
                              void* d_out, int out_size, void* d_ws, size_t ws_size,
                              hipStream_t stream) {
    (void)in_sizes; (void)n_in; (void)out_size; (void)ws_size;
    const float* in0 = (const float*)d_in[0];
    const float* in1 = (const float*)d_in[1];
    const float* in2 = (const float*)d_in[2];
    const float* in3 = (const float*)d_in[3];

    unsigned int* hist = (unsigned int*)d_ws;                    // 4*8*65536 u32 = 8 MiB
    const int nhist = NIN * BATCH * NB * NB;
    float* partials = (float*)((char*)d_ws + (size_t)nhist * sizeof(unsigned int));

    zero_hist_kernel<<<1024, 256, 0, stream>>>(hist, nhist);

    dim3 grid(BATCH * WPI, NIN);                                 // 128 x 4 workgroups
    hist_kernel<<<grid, 256, 0, stream>>>(in0, in1, in2, in3, hist);

    moment_kernel<<<NIN * BATCH * 16, 32, 0, stream>>>(hist, partials);

    final_kernel<<<1, 1, 0, stream>>>(partials, (float*)d_out);
}